// COMSAGPool_multi_scores_89060441850428
// MI455X (gfx1250) — compile-verified
//
#include <hip/hip_runtime.h>
#include <math.h>

// Problem constants (match reference)
#define BG   64
#define NN   1024
#define DD   256
#define NT   (BG * NN)
#define KK   512
#define NCOM (NT - BG * KK)

typedef float v2f __attribute__((ext_vector_type(2)));
typedef float v8f __attribute__((ext_vector_type(8)));

// ---------------- workspace layout (float offsets) ----------------
#define WS_GEMB    0            // B*D = 16384
#define WS_C0      16384        // 256
#define WS_C1      16640        // 256
#define WS_STATS   16896        // 16: [0]=n0 [1]=n1 [2]=cc0 [3]=cc1 [4]=gmax [5]=sumexp
#define WS_BMAT    16912        // D*16 = 4096
#define WS_FW      32768        // NT
#define WS_FC0     (WS_FW  + NT)
#define WS_FC1     (WS_FC0 + NT)
#define WS_SQ      (WS_FC1 + NT)
#define WS_DEGO    (WS_SQ  + NT)
#define WS_DEGI    (WS_DEGO + NT)
#define WS_H       (WS_DEGI + NT)
#define WS_SDIST   (WS_H   + NT)
#define WS_AGG     (WS_SDIST + NT)
#define WS_SFLAT   (WS_AGG + NT)
#define WS_PERM    (WS_SFLAT + NT)          // int, B*K
#define WS_PERMCOM (WS_PERM + BG * KK)      // int, NCOM

// ---------------- output layout (float offsets) -------------------
#define OUT_DIS     0
#define OUT_COM     (BG * KK * DD)
#define OUT_PERM    (OUT_COM + NCOM * DD)
#define OUT_PERMCOM (OUT_PERM + BG * KK)
#define OUT_SOFT    (OUT_PERMCOM + NCOM)

// =================================================================
__global__ __launch_bounds__(256) void init_kernel(float* degO, float* degI,
                                                   float* agg, float* stats) {
    int i = blockIdx.x * blockDim.x + threadIdx.x;
    if (i < NT) { degO[i] = 0.f; degI[i] = 0.f; agg[i] = 0.f; }
    if (i == 0) { stats[4] = -INFINITY; stats[5] = 0.f; }
}

// per-graph mean of features: one block per graph, thread = dim
__global__ __launch_bounds__(256) void gemb_kernel(const float* __restrict__ feat,
                                                   float* __restrict__ gemb) {
    int g = blockIdx.x, d = threadIdx.x;
    const float* base = feat + (size_t)g * NN * DD + d;
    float s = 0.f;
    for (int n = 0; n < NN; ++n) s += base[(size_t)n * DD];
    gemb[g * DD + d] = s * (1.0f / NN);
}

// centers + ||c||^2 + packed B-matrix [W | c0 | c1 | 0...]
__global__ __launch_bounds__(256) void centers_kernel(const float* __restrict__ gemb,
                                                      const int* __restrict__ label,
                                                      const float* __restrict__ W,
                                                      float* c0, float* c1,
                                                      float* stats, float* Bmat) {
    int d = threadIdx.x;
    float s0 = 0.f, s1 = 0.f;
    for (int g = 0; g < BG; ++g) {
        float v = gemb[g * DD + d];
        if (label[g]) s1 += v; else s0 += v;
    }
    __shared__ float n0s, n1s;
    if (d == 0) {
        int cnt1 = 0;
        for (int g = 0; g < BG; ++g) cnt1 += (label[g] ? 1 : 0);
        n1s = fmaxf((float)cnt1, 1.f);
        n0s = fmaxf((float)(BG - cnt1), 1.f);
        stats[0] = n0s; stats[1] = n1s;
    }
    __syncthreads();
    float C0 = s0 / n0s, C1 = s1 / n1s;
    c0[d] = C0; c1[d] = C1;

    __shared__ float r0[256], r1[256];
    r0[d] = C0 * C0; r1[d] = C1 * C1;
    __syncthreads();
    for (int s = 128; s > 0; s >>= 1) {
        if (d < s) { r0[d] += r0[d + s]; r1[d] += r1[d + s]; }
        __syncthreads();
    }
    if (d == 0) { stats[2] = r0[0]; stats[3] = r1[0]; }

    Bmat[d * 16 + 0] = W[d];
    Bmat[d * 16 + 1] = C0;
    Bmat[d * 16 + 2] = C1;
    #pragma unroll
    for (int n = 3; n < 16; ++n) Bmat[d * 16 + n] = 0.f;
}

// degree counting over edges
__global__ __launch_bounds__(256) void deg_kernel(const int* __restrict__ src,
                                                  const int* __restrict__ dst,
                                                  float* degO, float* degI, int E) {
    int e = blockIdx.x * blockDim.x + threadIdx.x;
    if (e >= E) return;
    atomicAdd(&degO[src[e]], 1.0f);
    atomicAdd(&degI[dst[e]], 1.0f);
}

// =================================================================
// Fused GEMV x3 via V_WMMA_F32_16X16X4_F32: one wave handles 16 nodes,
// D = 16x16 fp32 accumulator, cols 0/1/2 = f.W, f.c0, f.c1.
// A (16x4 f32): lanes 0-15 hold K=0,1 ; lanes 16-31 hold K=2,3 (ISA layout).
// ||f||^2 computed from the same A loads + shfl_xor(16).
__global__ __launch_bounds__(32) void wmma_dot_kernel(const float* __restrict__ feat,
                                                      const float* __restrict__ Bmat,
                                                      float* __restrict__ fW,
                                                      float* __restrict__ fc0,
                                                      float* __restrict__ fc1,
                                                      float* __restrict__ sq) {
    int base = blockIdx.x * 16;
    int lane = threadIdx.x;           // 0..31, one wave, EXEC all ones
    int m    = lane & 15;             // node row / B column
    int koff = (lane >> 4) * 2;       // 0 for lanes 0-15, 2 for lanes 16-31

    const float* frow = feat + (size_t)(base + m) * DD;
    v8f c = {};
    float s = 0.f;

    for (int kk = 0; kk < DD / 4; ++kk) {
        int k0 = kk * 4 + koff;
        v2f a;
        a.x = frow[k0];
        a.y = frow[k0 + 1];
        v2f b;
        b.x = Bmat[(size_t)k0 * 16 + m];
        b.y = Bmat[(size_t)(k0 + 1) * 16 + m];
        c = __builtin_amdgcn_wmma_f32_16x16x4_f32(
                /*neg_a=*/false, a, /*neg_b=*/false, b,
                /*c_mod=*/(short)0, c, /*reuse_a=*/false, /*reuse_b=*/false);
        s += a.x * a.x + a.y * a.y;
    }

    // lanes l and l+16 hold complementary K-halves of node base+l
    s += __shfl_xor(s, 16, 32);
    if (lane < 16) sq[base + lane] = s;

    // C/D layout: VGPR v -> M = v + 8*(lane>=16), N = lane&15
    int n    = lane & 15;
    int mrow = (lane >> 4) * 8;
    if (n < 3) {
        float* dstp = (n == 0) ? fW : ((n == 1) ? fc0 : fc1);
        #pragma unroll
        for (int v = 0; v < 8; ++v) dstp[base + mrow + v] = c[v];
    }
}

// =================================================================
__global__ __launch_bounds__(256) void node_score_kernel(const float* __restrict__ fW,
                                                         const float* __restrict__ fc0,
                                                         const float* __restrict__ fc1,
                                                         const float* __restrict__ sq,
                                                         const float* __restrict__ degO,
                                                         const int* __restrict__ label,
                                                         const float* __restrict__ stats,
                                                         float* __restrict__ h,
                                                         float* __restrict__ sdist) {
    int i = blockIdx.x * blockDim.x + threadIdx.x;
    if (i >= NT) return;
    int g = i >> 10;                       // N = 1024
    float q  = sq[i];
    float d0 = sqrtf(fmaxf(q - 2.f * fc0[i] + stats[2], 0.f));
    float d1 = sqrtf(fmaxf(q - 2.f * fc1[i] + stats[3], 0.f));
    float sgn = label[g] ? 1.f : -1.f;     // (label - 0.5) * 2
    sdist[i] = (d0 - d1) * sgn;
    float ns = rsqrtf(fmaxf(degO[i], 1.f));
    h[i] = fW[i] * ns;
}

__global__ __launch_bounds__(256) void edge_agg_kernel(const int* __restrict__ src,
                                                       const int* __restrict__ dst,
                                                       const float* __restrict__ h,
                                                       float* agg, int E) {
    int e = blockIdx.x * blockDim.x + threadIdx.x;
    if (e >= E) return;
    atomicAdd(&agg[dst[e]], h[src[e]]);
}

__device__ __forceinline__ void atomicMaxF(float* addr, float val) {
    unsigned* ua = (unsigned*)addr;
    unsigned old = *ua;
    while (true) {
        float of = __uint_as_float(old);
        if (of >= val) break;
        unsigned assumed = old;
        old = atomicCAS(ua, assumed, __float_as_uint(val));
        if (old == assumed) break;
    }
}

__global__ __launch_bounds__(256) void score_kernel(const float* __restrict__ agg,
                                                    const float* __restrict__ degI,
                                                    const float* __restrict__ sdist,
                                                    const float* __restrict__ b,
                                                    const float* __restrict__ lin1_w,
                                                    const float* __restrict__ lin1_b,
                                                    float* __restrict__ sflat,
                                                    float* stats) {
    int i = blockIdx.x * blockDim.x + threadIdx.x;
    float nd = rsqrtf(fmaxf(degI[i], 1.f));
    float sg = agg[i] * nd + b[0];
    float s  = sg * lin1_w[0] + sdist[i] * lin1_w[1] + lin1_b[0];
    sflat[i] = s;
    __shared__ float red[256];
    int t = threadIdx.x;
    red[t] = s;
    __syncthreads();
    for (int k = 128; k > 0; k >>= 1) {
        if (t < k) red[t] = fmaxf(red[t], red[t + k]);
        __syncthreads();
    }
    if (t == 0) atomicMaxF(&stats[4], red[0]);
}

__global__ __launch_bounds__(256) void sumexp_kernel(const float* __restrict__ sflat,
                                                     float* stats) {
    int i = blockIdx.x * blockDim.x + threadIdx.x;
    float e = expf(sflat[i] - stats[4]);
    __shared__ float red[256];
    int t = threadIdx.x;
    red[t] = e;
    __syncthreads();
    for (int k = 128; k > 0; k >>= 1) {
        if (t < k) red[t] += red[t + k];
        __syncthreads();
    }
    if (t == 0) atomicAdd(&stats[5], red[0]);
}

__global__ __launch_bounds__(256) void softmax_out_kernel(const float* __restrict__ sflat,
                                                          const float* __restrict__ stats,
                                                          float* __restrict__ out_soft) {
    int i = blockIdx.x * blockDim.x + threadIdx.x;
    out_soft[i] = expf(sflat[i] - stats[4]) / stats[5];
}

// exact top-K per graph with JAX tie-break (value desc, index asc),
// complement in ascending index order
__global__ __launch_bounds__(1024) void topk_kernel(const float* __restrict__ sflat,
                                                    int* __restrict__ perm,
                                                    int* __restrict__ permcom) {
    int g = blockIdx.x, i = threadIdx.x;
    __shared__ float sv[NN];
    __shared__ unsigned char sel[NN];
    float v = sflat[g * NN + i];
    sv[i] = v;
    __syncthreads();
    int rank = 0;
    for (int j = 0; j < NN; ++j) {
        float u = sv[j];
        rank += (u > v) || (u == v && j < i);
    }
    if (rank < KK) { perm[g * KK + rank] = g * NN + i; sel[i] = 1; }
    else            sel[i] = 0;
    __syncthreads();
    if (rank >= KK) {
        int pos = 0;
        for (int j = 0; j < i; ++j) pos += (sel[j] == 0);
        permcom[g * (NN - KK) + pos] = g * NN + i;
    }
}

// gated gathers: one block per output row, float4 path (D=256 -> 64 lanes)
__global__ __launch_bounds__(64) void gather_kernel(const float* __restrict__ feat,
                                                    const float* __restrict__ sflat,
                                                    const int* __restrict__ idx,
                                                    float* __restrict__ out_rows,
                                                    float* __restrict__ out_idx) {
    int p = blockIdx.x;
    int node = idx[p];
    float t = tanhf(sflat[node]);
    const float4* s = (const float4*)(feat + (size_t)node * DD);
    float4* d = (float4*)(out_rows + (size_t)p * DD);
    float4 v = s[threadIdx.x];
    v.x *= t; v.y *= t; v.z *= t; v.w *= t;
    d[threadIdx.x] = v;
    if (threadIdx.x == 0) out_idx[p] = (float)node;
}

// =================================================================
extern "C" void kernel_launch(void* const* d_in, const int* in_sizes, int n_in,
                              void* d_out, int out_size, void* d_ws, size_t ws_size,
                              hipStream_t stream) {
    const float* feature = (const float*)d_in[0];
    const int*   src     = (const int*)d_in[1];
    const int*   dst     = (const int*)d_in[2];
    const int*   label   = (const int*)d_in[3];
    const float* W       = (const float*)d_in[4];
    const float* b       = (const float*)d_in[5];
    const float* lin1_w  = (const float*)d_in[6];
    const float* lin1_b  = (const float*)d_in[7];
    const int E = in_sizes[1];

    float* ws  = (float*)d_ws;
    float* out = (float*)d_out;

    float* gemb  = ws + WS_GEMB;
    float* c0    = ws + WS_C0;
    float* c1    = ws + WS_C1;
    float* stats = ws + WS_STATS;
    float* Bmat  = ws + WS_BMAT;
    float* fW    = ws + WS_FW;
    float* fc0   = ws + WS_FC0;
    float* fc1   = ws + WS_FC1;
    float* sq    = ws + WS_SQ;
    float* degO  = ws + WS_DEGO;
    float* degI  = ws + WS_DEGI;
    float* h     = ws + WS_H;
    float* sdist = ws + WS_SDIST;
    float* agg   = ws + WS_AGG;
    float* sflat = ws + WS_SFLAT;
    int*   perm    = (int*)(ws + WS_PERM);
    int*   permcom = (int*)(ws + WS_PERMCOM);

    init_kernel<<<NT / 256, 256, 0, stream>>>(degO, degI, agg, stats);
    gemb_kernel<<<BG, 256, 0, stream>>>(feature, gemb);
    centers_kernel<<<1, 256, 0, stream>>>(gemb, label, W, c0, c1, stats, Bmat);
    deg_kernel<<<(E + 255) / 256, 256, 0, stream>>>(src, dst, degO, degI, E);
    wmma_dot_kernel<<<NT / 16, 32, 0, stream>>>(feature, Bmat, fW, fc0, fc1, sq);
    node_score_kernel<<<NT / 256, 256, 0, stream>>>(fW, fc0, fc1, sq, degO, label,
                                                    stats, h, sdist);
    edge_agg_kernel<<<(E + 255) / 256, 256, 0, stream>>>(src, dst, h, agg, E);
    score_kernel<<<NT / 256, 256, 0, stream>>>(agg, degI, sdist, b, lin1_w, lin1_b,
                                               sflat, stats);
    sumexp_kernel<<<NT / 256, 256, 0, stream>>>(sflat, stats);
    topk_kernel<<<BG, 1024, 0, stream>>>(sflat, perm, permcom);
    gather_kernel<<<BG * KK, 64, 0, stream>>>(feature, sflat, perm,
                                              out + OUT_DIS, out + OUT_PERM);
    gather_kernel<<<NCOM, 64, 0, stream>>>(feature, sflat, permcom,
                                           out + OUT_COM, out + OUT_PERMCOM);
    softmax_out_kernel<<<NT / 256, 256, 0, stream>>>(sflat, stats, out + OUT_SOFT);
}